// SAGPoolReadout_27565100105879
// MI455X (gfx1250) — compile-verified
//
#include <hip/hip_runtime.h>
#include <math.h>

#define G    64
#define N0   1024
#define D    128
#define NE   16384
#define NOUT 64

typedef __attribute__((ext_vector_type(2))) float v2f;
typedef __attribute__((ext_vector_type(8))) float v8f;

static inline int cdiv_i(int a, int b) { return (a + b - 1) / b; }

// ---------------- zero ----------------
__global__ void sag_zero_f32(float* __restrict__ p, int n) {
  int t = blockIdx.x * blockDim.x + threadIdx.x;
  if (t < n) p[t] = 0.0f;
}

// ------------- per-node dot products: xw_rel = x.w_rel, xw_root = x.w_root -------------
__global__ void sag_node_dots(const float* __restrict__ x,
                              const float* __restrict__ wrel,
                              const float* __restrict__ wroot,
                              float* __restrict__ xwrel,
                              float* __restrict__ xwroot,
                              int nNodes) {
  int wave = (blockIdx.x * blockDim.x + threadIdx.x) >> 5;
  int lane = threadIdx.x & 31;
  if (wave >= G * nNodes) return;
  const float* xp = x + (size_t)wave * D;
  float ar = 0.f, ao = 0.f;
#pragma unroll
  for (int i = 0; i < D; i += 32) {
    float v = xp[i + lane];
    ar += v * wrel[i + lane];
    ao += v * wroot[i + lane];
  }
#pragma unroll
  for (int m = 16; m >= 1; m >>= 1) {
    ar += __shfl_xor(ar, m, 32);
    ao += __shfl_xor(ao, m, 32);
  }
  if (lane == 0) { xwrel[wave] = ar; xwroot[wave] = ao; }
}

// ------------- scalar scatter-add: aggw[dst] += xw_rel[src] -------------
__global__ void sag_scatter_add(const int* __restrict__ ei,
                                const int* __restrict__ mask, // nullptr => all-true
                                const float* __restrict__ xw,
                                float* __restrict__ aggw,
                                int nNodes) {
  int t = blockIdx.x * blockDim.x + threadIdx.x;
  if (t >= G * NE) return;
  int g = t / NE, e = t - g * NE;
  if (mask && !mask[t]) return;
  int src = ei[(size_t)g * 2 * NE + e];
  int dst = ei[(size_t)g * 2 * NE + NE + e];
  atomicAdd(&aggw[g * nNodes + dst], xw[g * nNodes + src]);
}

// ------------- score = tanh((aggw + b_rel + xw_root) * sign(w_sel)) -------------
__global__ void sag_score(const float* __restrict__ aggw,
                          const float* __restrict__ xwroot,
                          const float* __restrict__ b_rel_p,
                          const float* __restrict__ w_sel_p,
                          float* __restrict__ score,
                          int total) {
  int t = blockIdx.x * blockDim.x + threadIdx.x;
  if (t >= total) return;
  float sgn = copysignf(1.0f, *w_sel_p);
  score[t] = tanhf((aggw[t] + *b_rel_p + xwroot[t]) * sgn);
}

// ------------- exact top-k rank per graph (stable, descending) -------------
__global__ void sag_rank(const float* __restrict__ score,
                         int* __restrict__ nidx,
                         int n, int k) {
  __shared__ float s[N0];
  int g = blockIdx.x;
  const float* sc = score + (size_t)g * n;
  for (int i = threadIdx.x; i < n; i += blockDim.x) s[i] = sc[i];
  __syncthreads();
  for (int node = threadIdx.x; node < n; node += blockDim.x) {
    float mine = s[node];
    int r = 0;
    for (int j = 0; j < n; ++j) {
      float v = s[j];
      r += (v > mine) || (v == mine && j < node);
    }
    nidx[(size_t)g * n + node] = (r < k) ? r : -1;
  }
}

// ------------- new_x[rank] = x[node] * score[node] -------------
__global__ void sag_gather(const float* __restrict__ x,
                           const float* __restrict__ score,
                           const int* __restrict__ nidx,
                           float* __restrict__ xout,
                           int n, int k) {
  int wave = (blockIdx.x * blockDim.x + threadIdx.x) >> 5;
  int lane = threadIdx.x & 31;
  if (wave >= G * n) return;
  int r = nidx[wave];
  if (r < 0) return;
  int g = wave / n;
  float sc = score[wave];
  const float* xp = x + (size_t)wave * D;
  float* op = xout + ((size_t)g * k + r) * D;
#pragma unroll
  for (int i = 0; i < D; i += 32) op[i + lane] = xp[i + lane] * sc;
}

// ------------- edge re-index + mask update -------------
__global__ void sag_reindex(const int* __restrict__ ei_in,
                            const int* __restrict__ mask_in, // nullptr => all-true
                            const int* __restrict__ nidx,
                            int n,
                            int* __restrict__ ei_out,
                            int* __restrict__ mask_out) {
  int t = blockIdx.x * blockDim.x + threadIdx.x;
  if (t >= G * NE) return;
  int g = t / NE, e = t - g * NE;
  int src = ei_in[(size_t)g * 2 * NE + e];
  int dst = ei_in[(size_t)g * 2 * NE + NE + e];
  int m = mask_in ? mask_in[t] : 1;
  int s2 = nidx[(size_t)g * n + src];
  int d2 = nidx[(size_t)g * n + dst];
  int nm = (m && s2 >= 0 && d2 >= 0) ? 1 : 0;
  ei_out[(size_t)g * 2 * NE + e] = nm ? s2 : 0;
  ei_out[(size_t)g * 2 * NE + NE + e] = nm ? d2 : 0;
  mask_out[t] = nm;
}

// ------------- readout: [mean(256 nodes), max(256 nodes)] per graph -------------
__global__ void sag_readout(const float* __restrict__ x2, float* __restrict__ ro) {
  int t = blockIdx.x * blockDim.x + threadIdx.x; // G*D threads
  if (t >= G * D) return;
  int g = t / D, d = t - g * D;
  const float* xp = x2 + (size_t)g * 256 * D + d;
  float sum = 0.f, mx = -INFINITY;
  for (int nn = 0; nn < 256; ++nn) {
    float v = xp[(size_t)nn * D];
    sum += v;
    mx = fmaxf(mx, v);
  }
  ro[(size_t)g * 256 + d] = sum * (1.0f / 256.0f);
  ro[(size_t)g * 256 + 128 + d] = mx;
}

// ------------- final head: (64x256)@(256x64)+b via V_WMMA_F32_16X16X4_F32 -------------
// One wave per 16x16 C tile. f32 WMMA layouts (ISA 7.12.2):
//   A 16x4:  lane L holds A[L&15][2*(L>>4) + {v0,v1}]
//   B 4x16:  lane L holds B[2*(L>>4) + {v0,v1}][L&15]
//   C 16x16: vgpr v, lane L -> C[v + 8*(L>>4)][L&15]
__global__ void __launch_bounds__(32)
sag_final_gemm(const float* __restrict__ R,   // 64 x 256
               const float* __restrict__ W,   // 256 x 64
               const float* __restrict__ bias,// 64
               float* __restrict__ out) {     // 64 x 64
  int mt = blockIdx.x * 16;
  int nt = blockIdx.y * 16;
  int lane = threadIdx.x;
  int half = lane >> 4;
  int l15 = lane & 15;
  v8f c = {};
  for (int k = 0; k < 2 * D; k += 4) {
    int ka = k + half * 2;
    v2f a, b;
    a.x = R[(size_t)(mt + l15) * 256 + ka];
    a.y = R[(size_t)(mt + l15) * 256 + ka + 1];
    b.x = W[(size_t)ka * NOUT + nt + l15];
    b.y = W[(size_t)(ka + 1) * NOUT + nt + l15];
    c = __builtin_amdgcn_wmma_f32_16x16x4_f32(
        /*neg_a=*/false, a, /*neg_b=*/false, b,
        /*c_mod=*/(short)0, c, /*reuse_a=*/false, /*reuse_b=*/false);
  }
  float bv = bias[nt + l15];
#pragma unroll
  for (int v = 0; v < 8; ++v) {
    out[(size_t)(mt + v + half * 8) * NOUT + nt + l15] = c[v] + bv;
  }
}

extern "C" void kernel_launch(void* const* d_in, const int* in_sizes, int n_in,
                              void* d_out, int out_size, void* d_ws, size_t ws_size,
                              hipStream_t stream) {
  const float* x0    = (const float*)d_in[0];  // G x N0 x D
  const int*   ei0   = (const int*)d_in[1];    // G x 2 x NE
  const float* w_rel = (const float*)d_in[2];  // L x D
  const float* b_rel = (const float*)d_in[3];  // L
  const float* w_root= (const float*)d_in[4];  // L x D
  const float* w_sel = (const float*)d_in[5];  // L
  const float* W_lin = (const float*)d_in[6];  // 2D x NOUT
  const float* b_lin = (const float*)d_in[7];  // NOUT
  float* out = (float*)d_out;                  // G x NOUT

  // -------- workspace layout --------
  char* ws = (char*)d_ws;
  size_t off = 0;
  auto take = [&](size_t bytes) { char* p = ws + off; off = (off + bytes + 255) & ~(size_t)255; return p; };
  float* x1     = (float*)take((size_t)G * 512 * D * 4); // after layer 1
  float* x2     = (float*)take((size_t)G * 256 * D * 4); // after layer 2
  float* xwrel  = (float*)take((size_t)G * N0 * 4);
  float* xwroot = (float*)take((size_t)G * N0 * 4);
  float* aggw   = (float*)take((size_t)G * N0 * 4);
  float* score  = (float*)take((size_t)G * N0 * 4);
  int*   nidx   = (int*)  take((size_t)G * N0 * 4);
  int*   ei1    = (int*)  take((size_t)G * 2 * NE * 4);
  int*   mask1  = (int*)  take((size_t)G * NE * 4);
  float* ro     = (float*)take((size_t)G * 256 * 4);

  const int BT = 256;

  // ================= Layer 1: n=1024, k=512 =================
  {
    int n = N0, k = 512;
    sag_node_dots<<<cdiv_i(G * n * 32, BT), BT, 0, stream>>>(x0, w_rel, w_root, xwrel, xwroot, n);
    sag_zero_f32<<<cdiv_i(G * n, BT), BT, 0, stream>>>(aggw, G * n);
    sag_scatter_add<<<cdiv_i(G * NE, BT), BT, 0, stream>>>(ei0, nullptr, xwrel, aggw, n);
    sag_score<<<cdiv_i(G * n, BT), BT, 0, stream>>>(aggw, xwroot, b_rel, w_sel, score, G * n);
    sag_rank<<<G, BT, 0, stream>>>(score, nidx, n, k);
    sag_gather<<<cdiv_i(G * n * 32, BT), BT, 0, stream>>>(x0, score, nidx, x1, n, k);
    sag_reindex<<<cdiv_i(G * NE, BT), BT, 0, stream>>>(ei0, nullptr, nidx, n, ei1, mask1);
  }

  // ================= Layer 2: n=512, k=256 =================
  {
    int n = 512, k = 256;
    sag_node_dots<<<cdiv_i(G * n * 32, BT), BT, 0, stream>>>(x1, w_rel + D, w_root + D, xwrel, xwroot, n);
    sag_zero_f32<<<cdiv_i(G * n, BT), BT, 0, stream>>>(aggw, G * n);
    sag_scatter_add<<<cdiv_i(G * NE, BT), BT, 0, stream>>>(ei1, mask1, xwrel, aggw, n);
    sag_score<<<cdiv_i(G * n, BT), BT, 0, stream>>>(aggw, xwroot, b_rel + 1, w_sel + 1, score, G * n);
    sag_rank<<<G, BT, 0, stream>>>(score, nidx, n, k);
    sag_gather<<<cdiv_i(G * n * 32, BT), BT, 0, stream>>>(x1, score, nidx, x2, n, k);
    // edges discarded after last layer
  }

  // ================= Readout + WMMA head =================
  sag_readout<<<cdiv_i(G * D, BT), BT, 0, stream>>>(x2, ro);
  sag_final_gemm<<<dim3(G / 16, NOUT / 16), 32, 0, stream>>>(ro, W_lin, b_lin, out);
}